// deeds_60842506715609
// MI455X (gfx1250) — compile-verified
//
#include <hip/hip_runtime.h>
#include <hip/hip_bf16.h>

// ---------------------------------------------------------------------------
// DEEDS registration pipeline for MI455X (gfx1250, wave32).
//   GS=16 -> g3=4096 control points, DW=15 -> 3375 displacements, C=8, VOL=64
// Stages:
//   K1 corr      : SSD correlation volume  -> ws0 [g][d]   (+ global_prefetch)
//   K2 minconv   : minpool3 + 2x avgpool3 over 15^3 disp cube (LDS frame 19^3)
//                  writes perm layout       -> ws1 [d][g]
//   K3 gridsmooth: pad2+avg3+avg3 over 16^3 grid == 16x16 matrix T applied
//                  along each axis, via V_WMMA_F32_16X16X4_F32 -> ws2 [g][d]
//   K4 combine   : a4 + a2*deeds + a3*avg (in place on ws2)
//   K2,K3 again  : second pass
//   K5 softmax   : softmax(-a5*c) + weighted displacement -> d_out
// ---------------------------------------------------------------------------

typedef float v2f __attribute__((ext_vector_type(2)));
typedef float v8f __attribute__((ext_vector_type(8)));

#define G3   4096
#define D3   3375
#define NTOT (13824000)   // 4096*3375
#define FR   19           // padded frame for displacement smoothing
#define FR3  6859         // 19^3

// ---------------------------------------------------------------------------
// K1: correlation volume. One block per control point g.
// ---------------------------------------------------------------------------
__global__ __launch_bounds__(256)
void corr_kernel(const float* __restrict__ feat00,
                 const float* __restrict__ feat50,
                 const float* __restrict__ grid_xyz,
                 const float* __restrict__ alpha,
                 float* __restrict__ deeds) {
    const int g = blockIdx.x;
    const int tid = threadIdx.x;
    __shared__ float sFix[8];
    __shared__ int   sI0[3][15];
    __shared__ int   sI1[3][15];
    __shared__ float sFr[3][15];

    const float gx = grid_xyz[g * 3 + 0];
    const float gy = grid_xyz[g * 3 + 1];
    const float gz = grid_xyz[g * 3 + 2];

    // Per-axis moving-sample tables (border clamp).
    if (tid < 45) {
        const int ax = tid / 15, i = tid % 15;
        const float gc = (ax == 0) ? gx : ((ax == 1) ? gy : gz);
        const float lin = -0.25f + (float)i * (0.5f / 14.0f);
        float p = (gc + lin + 1.0f) * 31.5f;
        p = fminf(fmaxf(p, 0.0f), 63.0f);
        const float pf = floorf(p);
        int i0 = (int)pf;
        int i1 = i0 + 1; if (i1 > 63) i1 = 63;
        sI0[ax][i] = i0; sI1[ax][i] = i1; sFr[ax][i] = p - pf;
    }
    // Fixed sample (zeros padding), one channel per thread.
    if (tid < 8) {
        const float x = (gx + 1.0f) * 31.5f;
        const float y = (gy + 1.0f) * 31.5f;
        const float z = (gz + 1.0f) * 31.5f;
        const float xf = floorf(x), yf = floorf(y), zf = floorf(z);
        const int x0 = (int)xf, y0 = (int)yf, z0 = (int)zf;
        const float tx = x - xf, ty = y - yf, tz = z - zf;
        const float* f = feat00 + (size_t)tid * 262144u;
        float acc = 0.0f;
        #pragma unroll
        for (int dz = 0; dz < 2; ++dz)
        #pragma unroll
        for (int dy = 0; dy < 2; ++dy)
        #pragma unroll
        for (int dx = 0; dx < 2; ++dx) {
            const int xi = x0 + dx, yi = y0 + dy, zi = z0 + dz;
            const float w = (dx ? tx : 1.0f - tx) * (dy ? ty : 1.0f - ty)
                          * (dz ? tz : 1.0f - tz);
            const bool valid = (xi >= 0) & (xi < 64) & (yi >= 0) & (yi < 64)
                             & (zi >= 0) & (zi < 64);
            const int cx = xi < 0 ? 0 : (xi > 63 ? 63 : xi);
            const int cy = yi < 0 ? 0 : (yi > 63 ? 63 : yi);
            const int cz = zi < 0 ? 0 : (zi > 63 ? 63 : zi);
            const float v = f[(cz * 64 + cy) * 64 + cx];
            acc += valid ? w * v : 0.0f;
        }
        sFix[tid] = acc;
    }
    __syncthreads();

    // Prefetch the (z,y) row starts of the sampled subvolume into cache
    // (global_prefetch_b8). 225 (z,y) pairs x 8 channels.
    if (tid < 225) {
        const int a = tid / 15, b = tid % 15;
        const int rowbase = (sI0[2][a] * 64 + sI0[1][b]) * 64 + sI0[0][0];
        #pragma unroll
        for (int ch = 0; ch < 8; ++ch)
            __builtin_prefetch(feat50 + (size_t)ch * 262144u + rowbase, 0, 1);
    }

    const float a0 = alpha[0], a1v = alpha[1];
    for (int d = tid; d < D3; d += 256) {
        const int a = d / 225, r = d - a * 225, b = r / 15, c = r - b * 15;
        const int ix0 = sI0[0][c], ix1 = sI1[0][c];
        const int iy0 = sI0[1][b], iy1 = sI1[1][b];
        const int iz0 = sI0[2][a], iz1 = sI1[2][a];
        const float fx = sFr[0][c], fy = sFr[1][b], fz = sFr[2][a];

        const int i000 = (iz0 * 64 + iy0) * 64 + ix0;
        const int i001 = (iz0 * 64 + iy0) * 64 + ix1;
        const int i010 = (iz0 * 64 + iy1) * 64 + ix0;
        const int i011 = (iz0 * 64 + iy1) * 64 + ix1;
        const int i100 = (iz1 * 64 + iy0) * 64 + ix0;
        const int i101 = (iz1 * 64 + iy0) * 64 + ix1;
        const int i110 = (iz1 * 64 + iy1) * 64 + ix0;
        const int i111 = (iz1 * 64 + iy1) * 64 + ix1;
        const float w000 = (1.0f - fz) * (1.0f - fy) * (1.0f - fx);
        const float w001 = (1.0f - fz) * (1.0f - fy) * fx;
        const float w010 = (1.0f - fz) * fy * (1.0f - fx);
        const float w011 = (1.0f - fz) * fy * fx;
        const float w100 = fz * (1.0f - fy) * (1.0f - fx);
        const float w101 = fz * (1.0f - fy) * fx;
        const float w110 = fz * fy * (1.0f - fx);
        const float w111 = fz * fy * fx;

        float ssd = 0.0f;
        #pragma unroll
        for (int ch = 0; ch < 8; ++ch) {
            const float* f = feat50 + (size_t)ch * 262144u;
            float m = w000 * f[i000] + w001 * f[i001]
                    + w010 * f[i010] + w011 * f[i011]
                    + w100 * f[i100] + w101 * f[i101]
                    + w110 * f[i110] + w111 * f[i111];
            const float df = sFix[ch] - m;
            ssd += df * df;
        }
        deeds[(size_t)g * D3 + d] = a1v + a0 * ssd;
    }
}

// ---------------------------------------------------------------------------
// K2: displacement-space min-convolution + smoothing. One block per g.
// Frame coords 0..18 map to original -2..16 (replicate pad via clamp).
// ---------------------------------------------------------------------------
__device__ __forceinline__ void axis_pass(const float* __restrict__ s,
                                          float* __restrict__ d,
                                          int axis, int isMin) {
    const int st = (axis == 0) ? 1 : ((axis == 1) ? FR : FR * FR);
    for (int p = threadIdx.x; p < FR3; p += 256) {
        const int z = p / (FR * FR);
        const int r = p - z * (FR * FR);
        const int y = r / FR;
        const int x = r - y * FR;
        const int c = (axis == 0) ? x : ((axis == 1) ? y : z);
        const int base = p - c * st;
        const int lo = (c - 1 < 0) ? 0 : c - 1;
        const int hi = (c + 1 > FR - 1) ? FR - 1 : c + 1;
        const float v0 = s[base + lo * st];
        const float v1 = s[p];
        const float v2 = s[base + hi * st];
        d[p] = isMin ? fminf(fminf(v0, v1), v2) : (v0 + v1 + v2) * (1.0f / 3.0f);
    }
}

__global__ __launch_bounds__(256)
void minconv_kernel(const float* __restrict__ in_g,    // [g][d]
                    float* __restrict__ out_perm) {    // [d][g]
    __shared__ float s0[FR3];
    __shared__ float s1[FR3];
    const int g = blockIdx.x;
    const int tid = threadIdx.x;
    const float* src = in_g + (size_t)g * D3;

    for (int p = tid; p < FR3; p += 256) {
        const int z = p / (FR * FR);
        const int r = p - z * (FR * FR);
        const int y = r / FR;
        const int x = r - y * FR;
        int oz = z - 2; oz = oz < 0 ? 0 : (oz > 14 ? 14 : oz);
        int oy = y - 2; oy = oy < 0 ? 0 : (oy > 14 ? 14 : oy);
        int ox = x - 2; ox = ox < 0 ? 0 : (ox > 14 ? 14 : ox);
        s0[p] = src[oz * 225 + oy * 15 + ox];
    }
    __syncthreads();
    axis_pass(s0, s1, 2, 1); __syncthreads();   // min z
    axis_pass(s1, s0, 1, 1); __syncthreads();   // min y
    axis_pass(s0, s1, 0, 1); __syncthreads();   // min x
    axis_pass(s1, s0, 2, 0); __syncthreads();   // avg1 z
    axis_pass(s0, s1, 1, 0); __syncthreads();   // avg1 y
    axis_pass(s1, s0, 0, 0); __syncthreads();   // avg1 x
    axis_pass(s0, s1, 2, 0); __syncthreads();   // avg2 z
    axis_pass(s1, s0, 1, 0); __syncthreads();   // avg2 y
    // final avg2 x, only the 15^3 interior (frame 2..16), write transposed.
    for (int o = tid; o < D3; o += 256) {
        const int a = o / 225, r = o - a * 225, b = r / 15, c = r - b * 15;
        const int p = (a + 2) * (FR * FR) + (b + 2) * FR + (c + 2);
        const float v = (s0[p - 1] + s0[p] + s0[p + 1]) * (1.0f / 3.0f);
        out_perm[(size_t)o * G3 + g] = v;
    }
}

// ---------------------------------------------------------------------------
// K3: grid-space smoothing via WMMA. One block (4 waves) per displacement d.
// The per-axis operator (replicate-pad 2, avg3, avg3) is a fixed 16x16 matrix
// T; the 3D operator is T applied along x, y, z of the 16^3 volume.
// A-fragment layout (ISA 7.12.2): lanes 0-15 hold K={4k,4k+1}, lanes 16-31
// hold K={4k+2,4k+3}, M = lane&15.  C/D: VGPR r -> M=r (lanes<16) / M=r+8.
// Tile loop counter is forced scalar (readfirstlane) so EXEC is statically
// all-1s at every WMMA (ISA 7.12 requirement) and the loop is pure SALU.
// ---------------------------------------------------------------------------
__global__ __launch_bounds__(128)
void grid_smooth_wmma(const float* __restrict__ in_perm,   // [d][g]
                      float* __restrict__ out_g) {         // [g][d]
    __shared__ float sT[256];    // T[i*16+j]
    __shared__ float sV[4096];
    __shared__ float sW[4096];
    const int d = blockIdx.x;
    const int tid = threadIdx.x;

    // Build T: T[i][j] = (1/9) * #{ (m,k) : m in [i,i+2], k in [m,m+2],
    //                               clamp(k-2,0,15)==j }
    for (int t = tid; t < 256; t += 128) {
        const int i = t >> 4, j = t & 15;
        int cnt = 0;
        for (int m = i; m <= i + 2; ++m)
            for (int k = m; k <= m + 2; ++k) {
                int o = k - 2; o = o < 0 ? 0 : (o > 15 ? 15 : o);
                cnt += (o == j);
            }
        sT[t] = (float)cnt * (1.0f / 9.0f);
    }
    for (int t = tid; t < 4096; t += 128)
        sV[t] = in_perm[(size_t)d * G3 + t];
    __syncthreads();

    const int lane  = tid & 31;
    const int wave  = __builtin_amdgcn_readfirstlane(tid >> 5);  // scalar
    const int l     = lane & 15;
    const int koff  = (lane >> 4) << 1;       // 0 | 2
    const int mbase = (lane >> 4) * 8;        // 0 | 8

    // ---- X pass: tile t is the z=t slice (rows y, cols x). Out = V * T^T.
    for (int t = wave; t < 16; t += 4) {
        v8f acc = {0.f, 0.f, 0.f, 0.f, 0.f, 0.f, 0.f, 0.f};
        #pragma unroll
        for (int kc = 0; kc < 4; ++kc) {
            const int k0 = kc * 4 + koff;
            v2f a, b;
            a.x = sV[(t * 16 + l) * 16 + k0];       // A[m=l][k]
            a.y = sV[(t * 16 + l) * 16 + k0 + 1];
            b.x = sT[l * 16 + k0];                  // B[k][n=l] = T[n][k]
            b.y = sT[l * 16 + k0 + 1];
            acc = __builtin_amdgcn_wmma_f32_16x16x4_f32(
                      false, a, false, b, (short)0, acc, false, false);
        }
        #pragma unroll
        for (int r = 0; r < 8; ++r)
            sW[(t * 16 + (mbase + r)) * 16 + l] = acc[r];
    }
    __syncthreads();

    // ---- Y pass: per z slice t. Out = T * W.
    for (int t = wave; t < 16; t += 4) {
        v8f acc = {0.f, 0.f, 0.f, 0.f, 0.f, 0.f, 0.f, 0.f};
        #pragma unroll
        for (int kc = 0; kc < 4; ++kc) {
            const int k0 = kc * 4 + koff;
            v2f a, b;
            a.x = sT[l * 16 + k0];                  // A[m=l][k] = T
            a.y = sT[l * 16 + k0 + 1];
            b.x = sW[(t * 16 + k0) * 16 + l];       // B[k][n=l]
            b.y = sW[(t * 16 + k0 + 1) * 16 + l];
            acc = __builtin_amdgcn_wmma_f32_16x16x4_f32(
                      false, a, false, b, (short)0, acc, false, false);
        }
        #pragma unroll
        for (int r = 0; r < 8; ++r)
            sV[(t * 16 + (mbase + r)) * 16 + l] = acc[r];
    }
    __syncthreads();

    // ---- Z pass: per y slice t. Out[z'][x] = sum_z T[z'][z] V[z][t][x].
    for (int t = wave; t < 16; t += 4) {
        v8f acc = {0.f, 0.f, 0.f, 0.f, 0.f, 0.f, 0.f, 0.f};
        #pragma unroll
        for (int kc = 0; kc < 4; ++kc) {
            const int k0 = kc * 4 + koff;
            v2f a, b;
            a.x = sT[l * 16 + k0];
            a.y = sT[l * 16 + k0 + 1];
            b.x = sV[(k0 * 16 + t) * 16 + l];
            b.y = sV[((k0 + 1) * 16 + t) * 16 + l];
            acc = __builtin_amdgcn_wmma_f32_16x16x4_f32(
                      false, a, false, b, (short)0, acc, false, false);
        }
        #pragma unroll
        for (int r = 0; r < 8; ++r) {
            const int gidx = (mbase + r) * 256 + t * 16 + l;   // z*256+y*16+x
            out_g[(size_t)gidx * D3 + d] = acc[r];
        }
    }
}

// ---------------------------------------------------------------------------
// K4: cost2 = a4 + a2*deeds + a3*cost_avg, in place on cost_avg (ws2).
// ---------------------------------------------------------------------------
__global__ __launch_bounds__(256)
void combine_kernel(const float* __restrict__ deeds,
                    float* __restrict__ ca,
                    const float* __restrict__ alpha) {
    const float a2 = alpha[2], a3 = alpha[3], a4 = alpha[4];
    for (int i = blockIdx.x * 256 + threadIdx.x; i < NTOT;
         i += gridDim.x * 256)
        ca[i] = a4 + a2 * deeds[i] + a3 * ca[i];
}

// ---------------------------------------------------------------------------
// K5: softmax over displacements + weighted displacement aggregation.
// ---------------------------------------------------------------------------
__global__ __launch_bounds__(256)
void softmax_pred_kernel(const float* __restrict__ cost,   // [g][d]
                         const float* __restrict__ shift,  // [d][3]
                         const float* __restrict__ alpha,
                         float* __restrict__ out) {
    const int g = blockIdx.x;
    const int tid = threadIdx.x;
    __shared__ float red[256];
    const float a5 = alpha[5];
    const float* cg = cost + (size_t)g * D3;

    float mx = -3.402823466e38f;
    for (int d = tid; d < D3; d += 256) mx = fmaxf(mx, -a5 * cg[d]);
    red[tid] = mx; __syncthreads();
    for (int s = 128; s > 0; s >>= 1) {
        if (tid < s) red[tid] = fmaxf(red[tid], red[tid + s]);
        __syncthreads();
    }
    mx = red[0]; __syncthreads();

    float sum = 0.0f;
    for (int d = tid; d < D3; d += 256) sum += __expf(-a5 * cg[d] - mx);
    red[tid] = sum; __syncthreads();
    for (int s = 128; s > 0; s >>= 1) {
        if (tid < s) red[tid] += red[tid + s];
        __syncthreads();
    }
    const float inv = 1.0f / red[0]; __syncthreads();

    float px = 0.0f, py = 0.0f, pz = 0.0f;
    for (int d = tid; d < D3; d += 256) {
        const float p = __expf(-a5 * cg[d] - mx) * inv;
        out[(size_t)g * D3 + d] = p;
        px += p * shift[d * 3 + 0];
        py += p * shift[d * 3 + 1];
        pz += p * shift[d * 3 + 2];
    }
    // three tree reductions
    red[tid] = px; __syncthreads();
    for (int s = 128; s > 0; s >>= 1) {
        if (tid < s) red[tid] += red[tid + s];
        __syncthreads();
    }
    px = red[0]; __syncthreads();
    red[tid] = py; __syncthreads();
    for (int s = 128; s > 0; s >>= 1) {
        if (tid < s) red[tid] += red[tid + s];
        __syncthreads();
    }
    py = red[0]; __syncthreads();
    red[tid] = pz; __syncthreads();
    for (int s = 128; s > 0; s >>= 1) {
        if (tid < s) red[tid] += red[tid + s];
        __syncthreads();
    }
    if (tid == 0) {
        float* pred = out + (size_t)NTOT + (size_t)g * 3;
        pred[0] = px; pred[1] = py; pred[2] = red[0];
    }
}

// ---------------------------------------------------------------------------
extern "C" void kernel_launch(void* const* d_in, const int* in_sizes, int n_in,
                              void* d_out, int out_size, void* d_ws,
                              size_t ws_size, hipStream_t stream) {
    const float* feat00 = (const float*)d_in[0];
    const float* feat50 = (const float*)d_in[1];
    const float* grid   = (const float*)d_in[2];
    // d_in[3] grid_size, d_in[4] displacement_width: compile-time constants.
    const float* shift  = (const float*)d_in[5];
    const float* alpha  = (const float*)d_in[6];
    float* out = (float*)d_out;

    float* ws0 = (float*)d_ws;          // deeds cost   [g][d]
    float* ws1 = ws0 + NTOT;            // perm buffer  [d][g]
    float* ws2 = ws1 + NTOT;            // g-major scratch / cost_avg

    corr_kernel<<<G3, 256, 0, stream>>>(feat00, feat50, grid, alpha, ws0);
    minconv_kernel<<<G3, 256, 0, stream>>>(ws0, ws1);
    grid_smooth_wmma<<<D3, 128, 0, stream>>>(ws1, ws2);
    combine_kernel<<<8192, 256, 0, stream>>>(ws0, ws2, alpha);
    minconv_kernel<<<G3, 256, 0, stream>>>(ws2, ws1);
    grid_smooth_wmma<<<D3, 128, 0, stream>>>(ws1, ws2);
    softmax_pred_kernel<<<G3, 256, 0, stream>>>(ws2, shift, alpha, out);
}